// DifferentiableVoronoiPropagation_37185826848907
// MI455X (gfx1250) — compile-verified
//
#include <hip/hip_runtime.h>
#include <stdint.h>

// Problem constants (match reference)
#define HH 224
#define WW 224
#define CC 196
#define HWSZ (HH * WW)
#define NUM_ITERS 20

#define ALPHAV  10.0f
#define BIGV    1.0e4f
#define GRADW   10.0f
#define COLORW  10.0f

// ---------------------------------------------------------------------------
// Kernel 1: per-pixel direction costs (iteration-invariant), packed AoS float4
//   ex/ey = Sobel (cross-correlation, zero-padded) on ITU-601 grayscale
//   wgrad = 10 * (ex^2+ey^2)^2  ==  10 * grad_map^4
//   costs4[p].{x,y,z,w} = wgrad + 10 * sum_ch |x[p] - x[wrapped nbr dir i]|
// ---------------------------------------------------------------------------
__global__ __launch_bounds__(256) void costs_kernel(
    const float* __restrict__ x, const float* __restrict__ ew,
    float4* __restrict__ costs4) {
  int p = blockIdx.x * blockDim.x + threadIdx.x;
  if (p >= HWSZ) return;
  int y = p / WW, xx = p - y * WW;

  float ex = 0.f, ey = 0.f;
#pragma unroll
  for (int ky = 0; ky < 3; ++ky) {
#pragma unroll
    for (int kx = 0; kx < 3; ++kx) {
      int yy = y + ky - 1, xc = xx + kx - 1;
      float g = 0.f;
      if (yy >= 0 && yy < HH && xc >= 0 && xc < WW) {
        int q = yy * WW + xc;
        g = 0.2989f * x[q] + 0.587f * x[q + HWSZ] + 0.114f * x[q + 2 * HWSZ];
      }
      ex += ew[ky * 3 + kx] * g;       // sobel_x (o=0)
      ey += ew[9 + ky * 3 + kx] * g;   // sobel_y (o=1)
    }
  }
  float s = ex * ex + ey * ey;
  float wgrad = GRADW * s * s;

  const int dy[4] = {-1, 1, 0, 0};
  const int dx[4] = {0, 0, -1, 1};
  float cv[4];
#pragma unroll
  for (int i = 0; i < 4; ++i) {
    // roll(x,(dy,dx))[y][x] = x[(y-dy) mod H][(x-dx) mod W]
    int ny = y - dy[i]; ny += (ny < 0) ? HH : 0; ny -= (ny >= HH) ? HH : 0;
    int nx = xx - dx[i]; nx += (nx < 0) ? WW : 0; nx -= (nx >= WW) ? WW : 0;
    int q = ny * WW + nx;
    float cd = fabsf(x[p] - x[q]) + fabsf(x[p + HWSZ] - x[q + HWSZ]) +
               fabsf(x[p + 2 * HWSZ] - x[q + 2 * HWSZ]);
    cv[i] = wgrad + COLORW * cd;
  }
  costs4[p] = make_float4(cv[0], cv[1], cv[2], cv[3]);
}

// ---------------------------------------------------------------------------
// Kernel 2: seed distance field: BIG everywhere, 0 at channel c's centroid
// ---------------------------------------------------------------------------
__global__ __launch_bounds__(256) void init_kernel(
    const int* __restrict__ cy, const int* __restrict__ cx,
    float* __restrict__ dist) {
  int idx = blockIdx.x * blockDim.x + threadIdx.x;
  if (idx >= CC * HWSZ) return;
  int c = idx / HWSZ;
  int p = idx - c * HWSZ;
  int y = p / WW, xx = p - y * WW;
  dist[idx] = (y == cy[c] && xx == cx[c]) ? 0.0f : BIGV;
}

// ---------------------------------------------------------------------------
// Kernel 3: one soft-min relaxation sweep, 4 channels per block.
// Stages CPB 32x8(+wrap-halo) channel tiles of dist into LDS via the CDNA5
// async global->LDS path (ASYNCcnt-tracked; per-lane scattered addresses
// implement the torus wrap), waits s_wait_asynccnt + barrier, then runs the
// 5-point stencil from LDS. The per-pixel float4 cost vector is loaded once
// (one b128) and reused across the 4 channels.
//   d_new = m - log( sum_k exp(-ALPHA*(cand_k - m)) ) / ALPHA,  m = min cand
// (identical to the reference's max-subtracted logsumexp)
// ---------------------------------------------------------------------------
#define TX 32
#define TY 8
#define LW (TX + 2)   // 34
#define LH (TY + 2)   // 10
#define TILEN (LH * LW)  // 340
#define CPB 4            // channels per block (196 = 49 * 4)

__global__ __launch_bounds__(256) void step_kernel(
    const float* __restrict__ src, const float4* __restrict__ costs4,
    float* __restrict__ dst) {
  __shared__ float tile[CPB][TILEN];

  const int cBase = blockIdx.z * CPB;
  const int x0 = blockIdx.x * TX;
  const int y0 = blockIdx.y * TY;
  const int tid = threadIdx.x;

  // LDS aperture puts the wave-relative LDS byte offset in addr[31:0]
  uint32_t ldsBase = (uint32_t)(uintptr_t)(void*)&tile[0][0];

  for (int i = tid; i < CPB * TILEN; i += 256) {
    int ch = i / TILEN;
    int j  = i - ch * TILEN;
    int ly = j / LW;
    int lx = j - ly * LW;
    int gy = y0 + ly - 1; gy = (gy < 0) ? gy + HH : ((gy >= HH) ? gy - HH : gy);
    int gx = x0 + lx - 1; gx = (gx < 0) ? gx + WW : ((gx >= WW) ? gx - WW : gx);
    const float* gp = src + (size_t)(cBase + ch) * HWSZ + gy * WW + gx;
    uint32_t la = ldsBase + (uint32_t)(i * 4);
    // GLOBAL_LOAD_ASYNC_TO_LDS_B32: vdst = per-lane LDS byte addr, vaddr = 64b global
    asm volatile("global_load_async_to_lds_b32 %0, %1, off"
                 :: "v"(la), "v"(gp) : "memory");
  }
  asm volatile("s_wait_asynccnt 0" ::: "memory");
  __syncthreads();

  const int lx = tid & (TX - 1);
  const int ly = tid >> 5;
  const int p  = (y0 + ly) * WW + (x0 + lx);

  const float4 cst = costs4[p];  // one b128, reused for all CPB channels

#pragma unroll
  for (int ch = 0; ch < CPB; ++ch) {
    float c0 = tile[ch][(ly + 1) * LW + (lx + 1)];
    float c1 = tile[ch][(ly + 2) * LW + (lx + 1)] + cst.x; // dir (-1,0)
    float c2 = tile[ch][(ly + 0) * LW + (lx + 1)] + cst.y; // dir ( 1,0)
    float c3 = tile[ch][(ly + 1) * LW + (lx + 2)] + cst.z; // dir (0,-1)
    float c4 = tile[ch][(ly + 1) * LW + (lx + 0)] + cst.w; // dir (0, 1)

    float m = fminf(fminf(fminf(c0, c1), fminf(c2, c3)), c4);
    float ssum = __expf(-ALPHAV * (c0 - m)) + __expf(-ALPHAV * (c1 - m)) +
                 __expf(-ALPHAV * (c2 - m)) + __expf(-ALPHAV * (c3 - m)) +
                 __expf(-ALPHAV * (c4 - m));
    dst[(size_t)(cBase + ch) * HWSZ + p] = m - __logf(ssum) * (1.0f / ALPHAV);
  }
}

// ---------------------------------------------------------------------------
// Kernel 4: channel softmax of -dist (min-shifted; L2-hot strided passes)
// ---------------------------------------------------------------------------
__global__ __launch_bounds__(256) void softmax_kernel(
    const float* __restrict__ dist, float* __restrict__ out) {
  int p = blockIdx.x * blockDim.x + threadIdx.x;
  if (p >= HWSZ) return;
  float m = dist[p];
  for (int c = 1; c < CC; ++c) m = fminf(m, dist[(size_t)c * HWSZ + p]);
  float S = 0.f;
  for (int c = 0; c < CC; ++c) S += __expf(m - dist[(size_t)c * HWSZ + p]);
  float inv = 1.0f / S;
  for (int c = 0; c < CC; ++c)
    out[(size_t)c * HWSZ + p] = __expf(m - dist[(size_t)c * HWSZ + p]) * inv;
}

// ---------------------------------------------------------------------------
extern "C" void kernel_launch(void* const* d_in, const int* in_sizes, int n_in,
                              void* d_out, int out_size, void* d_ws, size_t ws_size,
                              hipStream_t stream) {
  const float* x  = (const float*)d_in[0];  // (1,3,224,224)
  const float* ew = (const float*)d_in[1];  // (2,1,3,3)
  const int*   cy = (const int*)d_in[2];    // (196,)
  const int*   cx = (const int*)d_in[3];    // (196,)
  float* out = (float*)d_out;               // (1,196,224,224)

  // ws layout: [dist0: C*H*W f32 = 39.3MB][costs4: H*W float4 = 0.8MB]
  float*  dist0  = (float*)d_ws;
  float4* costs4 = (float4*)(dist0 + (size_t)CC * HWSZ);  // 16B-aligned (39,337,984 % 16 == 0)

  costs_kernel<<<(HWSZ + 255) / 256, 256, 0, stream>>>(x, ew, costs4);
  init_kernel<<<(CC * HWSZ + 255) / 256, 256, 0, stream>>>(cy, cx, dist0);

  dim3 grid(WW / TX, HH / TY, CC / CPB);  // (7, 28, 49)
  float* a = dist0;
  float* b = out;  // d_out doubles as the ping-pong partner
  for (int it = 0; it < NUM_ITERS; ++it) {
    step_kernel<<<grid, 256, 0, stream>>>(a, costs4, b);
    float* t = a; a = b; b = t;
  }
  // NUM_ITERS even -> final dist is back in ws (a == dist0)
  softmax_kernel<<<(HWSZ + 255) / 256, 256, 0, stream>>>(a, out);
}